// HolographicConv_41171556500230
// MI455X (gfx1250) — compile-verified
//
#include <hip/hip_runtime.h>

// ===========================================================================
// HolographicConv on MI455X (gfx1250):  out = real(ifft2(fft2(x) * w))
// B=16, C=16, H=W=256  (x, w complex, given as separate re/im f32 planes)
//
// Roofline: min HBM traffic = x(128MiB)+w(128MiB)+out(64MiB) ~= 320MiB
//           -> ~14us @ 23.3 TB/s.  FFT flops are small when each 256-pt
//           (i)FFT is done as the four-step 16x16 decomposition:
//              two 16x16 complex matmuls + one twiddle
//           which maps 1:1 onto V_WMMA_F32_16X16X4_F32 (K=16 via 4 chained
//           K=4 WMMAs; complex matmul = 4 real matmuls = 16 WMMAs/stage).
//           One wave32 owns one full 16x16 tile = one 256-vector transform.
//
// 3-kernel pipeline; the 128MiB float2 intermediate lives in d_ws and is
// essentially resident in MI455X's 192MB L2 between kernels, keeping HBM
// traffic near the 320MiB floor.
//   K1: FFT along W (rows),  natural-frequency-order output -> ws
//   K2: FFT along H (cols) -> *w -> iFFT along H, in-place in ws
//   K3: iFFT along W (rows), * 1/(H*W), real part -> d_out
//
// Assumes ws_size >= 16*16*256*256*sizeof(float2) = 128 MiB.
//
// Fragment layouts per CDNA5 ISA 7.12.2 (wave32, f32):
//   A 16x4 : lane L: row M = L&15; v0 = K(2*half), v1 = K(2*half+1)
//   B 4x16 : lane L: col N = L&15; v0 = K(2*half), v1 = K(2*half+1)
//   C/D    : VGPR v: row M = v + 8*half, col N = L&15
// F32 WMMA has no A/B NEG modifier (NEG = CNeg only), so -Ai*Bi uses a
// VALU-negated B fragment.
// ===========================================================================

typedef float v2f __attribute__((ext_vector_type(2)));
typedef float v8f __attribute__((ext_vector_type(8)));

#define WMMA_F32(a, b, c) \
  __builtin_amdgcn_wmma_f32_16x16x4_f32(false, (a), false, (b), (short)0, (c), false, false)

__device__ __forceinline__ float2 cmul2(float2 a, float2 b) {
  return make_float2(fmaf(a.x, b.x, -(a.y * b.y)), fmaf(a.x, b.y, a.y * b.x));
}

// ---- fragment loaders -----------------------------------------------------
// generic 16x16 complex matrix in LDS, element (r,c) at m[r*rs + c*cs]

__device__ __forceinline__ void fragA_mat(const float2* m, int rs, int cs, int kk,
                                          int l16, int half, v2f& ar, v2f& ai) {
  int c = 4 * kk + 2 * half;
  float2 e0 = m[l16 * rs + c * cs];
  float2 e1 = m[l16 * rs + (c + 1) * cs];
  ar[0] = e0.x; ar[1] = e1.x;
  ai[0] = e0.y; ai[1] = e1.y;
}

__device__ __forceinline__ void fragB_mat(const float2* m, int rs, int cs, int kk,
                                          int l16, int half, v2f& br, v2f& bi) {
  int r = 4 * kk + 2 * half;
  float2 e0 = m[r * rs + l16 * cs];
  float2 e1 = m[(r + 1) * rs + l16 * cs];
  br[0] = e0.x; br[1] = e1.x;
  bi[0] = e0.y; bi[1] = e1.y;
}

// F16[r][c] = w256^(-16*r*c) fetched from tw table (tw[t] = w256^(-t)).
// isgn = -1 conjugates (gives the inverse-DFT matrix G16).
__device__ __forceinline__ void fragA_F(const float2* tw, int kk, int l16, int half,
                                        float isgn, v2f& ar, v2f& ai) {
  int c = 4 * kk + 2 * half;
  float2 e0 = tw[(16 * l16 * c) & 255];
  float2 e1 = tw[(16 * l16 * (c + 1)) & 255];
  ar[0] = e0.x;        ar[1] = e1.x;
  ai[0] = isgn * e0.y; ai[1] = isgn * e1.y;
}

__device__ __forceinline__ void fragB_F(const float2* tw, int kk, int l16, int half,
                                        float isgn, v2f& br, v2f& bi) {
  int r = 4 * kk + 2 * half;
  float2 e0 = tw[(16 * r * l16) & 255];
  float2 e1 = tw[(16 * (r + 1) * l16) & 255];
  br[0] = e0.x;        br[1] = e1.x;
  bi[0] = isgn * e0.y; bi[1] = isgn * e1.y;
}

// complex MAC: (Dr,Di) += (Ar + i Ai)(Br + i Bi)  -> 4 WMMAs
__device__ __forceinline__ void cmacc(v2f ar, v2f ai, v2f br, v2f bi, v8f& dr, v8f& di) {
  v2f nbi = -bi;
  dr = WMMA_F32(ar, br, dr);
  dr = WMMA_F32(ai, nbi, dr);
  di = WMMA_F32(ar, bi, di);
  di = WMMA_F32(ai, br, di);
}

// ---- one 256-point transform per wave (two 16x16 complex WMMA matmuls) ----
// fwd (INV=false): vec natural-order in       -> vec natural-frequency out
// inv (INV=true) : vec natural-frequency in   -> vec natural-order out (unscaled)
//
// Derivation (n = 16*n1+n2, k = k1+16*k2):
//   fwd: D = (F16 . A) o T then . F16, A[n1][n2]=vec[16n1+n2],
//        T[k1][n2]=w256^(-k1*n2), D[k1][k2]=X[k1+16k2]  -> scatter to vec[k1+16k2]
//   inv: Out = G16 . ((Ymat . G16) o conj(T)), Ymat[k1][k2]=vec[k1+16k2],
//        Out[m1][m2]=y[16m1+m2] -> store vec[16m1+m2]
template <bool INV>
__device__ __forceinline__ void xform256(float2* vec, float2* scr,
                                         const float2* tw, int lane) {
  const int l16 = lane & 15, half = lane >> 4;
  const float isgn = INV ? -1.f : 1.f;
  const v8f zero = {0.f, 0.f, 0.f, 0.f, 0.f, 0.f, 0.f, 0.f};
  v8f dr = zero, di = zero;

  // ---- stage 1 ----
#pragma unroll
  for (int kk = 0; kk < 4; ++kk) {
    v2f ar, ai, br, bi;
    if (!INV) {  // F16 . A   : A-op = F16, B-op = data (row-major 16x16)
      fragA_F(tw, kk, l16, half, 1.f, ar, ai);
      fragB_mat(vec, 16, 1, kk, l16, half, br, bi);
    } else {     // Ymat . G16: A-op = data (digit-swapped gather), B-op = G16
      fragA_mat(vec, 1, 16, kk, l16, half, ar, ai);
      fragB_F(tw, kk, l16, half, -1.f, br, bi);
    }
    cmacc(ar, ai, br, bi, dr, di);
  }

  // ---- twiddle, D -> scratch matrix (row-major) ----
#pragma unroll
  for (int v = 0; v < 8; ++v) {
    int M = v + 8 * half, N = l16;
    float2 t = tw[(M * N) & 255];
    t.y *= isgn;
    scr[M * 16 + N] = cmul2(make_float2(dr[v], di[v]), t);
  }

  // ---- stage 2 ----
  dr = zero; di = zero;
#pragma unroll
  for (int kk = 0; kk < 4; ++kk) {
    v2f ar, ai, br, bi;
    if (!INV) {  // (.) . F16 : A-op = twiddled intermediate, B-op = F16
      fragA_mat(scr, 16, 1, kk, l16, half, ar, ai);
      fragB_F(tw, kk, l16, half, 1.f, br, bi);
    } else {     // G16 . (.) : A-op = G16, B-op = twiddled intermediate
      fragA_F(tw, kk, l16, half, -1.f, ar, ai);
      fragB_mat(scr, 16, 1, kk, l16, half, br, bi);
    }
    cmacc(ar, ai, br, bi, dr, di);
  }

  // ---- final scatter back to vec ----
#pragma unroll
  for (int v = 0; v < 8; ++v) {
    int M = v + 8 * half, N = l16;
    int p = INV ? (16 * M + N) : (M + 16 * N);  // natural order either way
    vec[p] = make_float2(dr[v], di[v]);
  }
}

__device__ __forceinline__ void build_tw(float2* s_tw, int tid) {
  // tw[t] = exp(-2*pi*i*t/256)
  float a = (float)tid * 0.02454369260617025967f;  // 2*pi/256
  float s, c;
  __sincosf(a, &s, &c);
  s_tw[tid] = make_float2(c, -s);
}

// ===========================================================================
// K1: forward FFT along W for every (b,c,h) row.  8 waves/block = 8 rows.
// ===========================================================================
__global__ __launch_bounds__(256) void hconv_k1_rowfft(
    const float* __restrict__ xr, const float* __restrict__ xi,
    float2* __restrict__ ws) {
  __shared__ float2 s_vec[8][256];
  __shared__ float2 s_scr[8][256];
  __shared__ float2 s_tw[256];
  const int tid = threadIdx.x, lane = tid & 31, wv = tid >> 5;
  build_tw(s_tw, tid);
  __syncthreads();

  const size_t base = ((size_t)blockIdx.x * 8 + wv) * 256;
#pragma unroll
  for (int j = 0; j < 8; ++j) {  // coalesced 128B/wave per plane
    int e = j * 32 + lane;
    s_vec[wv][e] = make_float2(xr[base + e], xi[base + e]);
  }
  xform256<false>(s_vec[wv], s_scr[wv], s_tw, lane);
#pragma unroll
  for (int j = 0; j < 8; ++j) {  // coalesced 256B/wave float2 store
    int e = j * 32 + lane;
    ws[base + e] = s_vec[wv][e];
  }
}

// ===========================================================================
// K2: per image, 8 columns/block: col-FFT -> * w -> col-iFFT (in-place ws).
//     Tile staged through LDS so global accesses stay coalesced.
// ===========================================================================
__global__ __launch_bounds__(256) void hconv_k2_colfft(
    float2* __restrict__ ws,
    const float* __restrict__ wr, const float* __restrict__ wi) {
  __shared__ float2 s_vec[8][256];
  __shared__ float2 s_scr[8][256];
  __shared__ float2 s_w[8][256];
  __shared__ float2 s_tw[256];
  const int tid = threadIdx.x, lane = tid & 31, wv = tid >> 5;
  build_tw(s_tw, tid);

  const int bc = blockIdx.x >> 5;            // image index (b*16+c)
  const int c0 = (blockIdx.x & 31) * 8;      // first column of tile
  const size_t ibase = (size_t)bc * 65536;

  // cooperative tile load: 8 consecutive threads hit 64B-contiguous global
#pragma unroll
  for (int j = 0; j < 8; ++j) {
    int e = tid + 256 * j;
    int col = e & 7, h = e >> 3;
    size_t g = ibase + (size_t)h * 256 + (c0 + col);
    s_vec[col][h] = ws[g];
    s_w[col][h] = make_float2(wr[g], wi[g]);
  }
  __syncthreads();

  // wave wv owns column (c0 + wv)
  xform256<false>(s_vec[wv], s_scr[wv], s_tw, lane);
#pragma unroll
  for (int j = 0; j < 8; ++j) {  // pointwise frequency-domain product
    int p = j * 32 + lane;
    s_vec[wv][p] = cmul2(s_vec[wv][p], s_w[wv][p]);
  }
  xform256<true>(s_vec[wv], s_scr[wv], s_tw, lane);
  __syncthreads();

#pragma unroll
  for (int j = 0; j < 8; ++j) {  // cooperative coalesced store back
    int e = tid + 256 * j;
    int col = e & 7, h = e >> 3;
    size_t g = ibase + (size_t)h * 256 + (c0 + col);
    ws[g] = s_vec[col][h];
  }
}

// ===========================================================================
// K3: inverse FFT along W, scale 1/(256*256), real part -> out.
// ===========================================================================
__global__ __launch_bounds__(256) void hconv_k3_irowfft(
    const float2* __restrict__ ws, float* __restrict__ out) {
  __shared__ float2 s_vec[8][256];
  __shared__ float2 s_scr[8][256];
  __shared__ float2 s_tw[256];
  const int tid = threadIdx.x, lane = tid & 31, wv = tid >> 5;
  build_tw(s_tw, tid);
  __syncthreads();

  const size_t base = ((size_t)blockIdx.x * 8 + wv) * 256;
#pragma unroll
  for (int j = 0; j < 8; ++j) {
    int e = j * 32 + lane;
    s_vec[wv][e] = ws[base + e];
  }
  xform256<true>(s_vec[wv], s_scr[wv], s_tw, lane);
  const float scale = 1.f / 65536.f;  // 1/(H*W) from ifft2
#pragma unroll
  for (int j = 0; j < 8; ++j) {
    int e = j * 32 + lane;
    out[base + e] = s_vec[wv][e].x * scale;
  }
}

// ===========================================================================
extern "C" void kernel_launch(void* const* d_in, const int* in_sizes, int n_in,
                              void* d_out, int out_size, void* d_ws, size_t ws_size,
                              hipStream_t stream) {
  (void)in_sizes; (void)n_in; (void)out_size; (void)ws_size;
  const float* xr = (const float*)d_in[0];
  const float* xi = (const float*)d_in[1];
  const float* wr = (const float*)d_in[2];
  const float* wi = (const float*)d_in[3];
  float2* ws = (float2*)d_ws;  // needs 128 MiB (fits MI455X 192MB L2)

  const int nvec_blocks = (16 * 16 * 256) / 8;  // 8192: one wave per 256-vector
  hconv_k1_rowfft<<<nvec_blocks, 256, 0, stream>>>(xr, xi, ws);
  hconv_k2_colfft<<<nvec_blocks, 256, 0, stream>>>(ws, wr, wi);
  hconv_k3_irowfft<<<nvec_blocks, 256, 0, stream>>>(ws, (float*)d_out);
}